// DR2FWL2Conv_88021059764495
// MI455X (gfx1250) — compile-verified
//
#include <hip/hip_runtime.h>

// fp32 WMMA fragments on gfx1250 (wave32):
//   A 16x4 f32  -> 2 VGPRs (v2f), B 4x16 f32 -> 2 VGPRs, C/D 16x16 f32 -> 8 VGPRs (v8f)
typedef __attribute__((ext_vector_type(2))) float v2f;
typedef __attribute__((ext_vector_type(8))) float v8f;

__device__ __forceinline__ v8f wmma_f32(v2f a, v2f b, v8f c) {
  // 8 args: (neg_a, A, neg_b, B, c_mod, C, reuse_a, reuse_b)
  return __builtin_amdgcn_wmma_f32_16x16x4_f32(false, a, false, b, (short)0, c,
                                               false, false);
}

// Cooperatively pack W[64x64] (row-major KxN, global) into LDS as v2f pairs:
//   Wl[(kk*2+kh)*64 + c] = ( W[k0][c], W[k0+1][c] ),  k0 = kk*4 + 2*kh
// so each B-fragment is a single aligned ds_load_b64.
__device__ __forceinline__ void pack_W(v2f* Wl, const float* __restrict__ W,
                                       int tid, int nthreads) {
  for (int i = tid; i < 32 * 64; i += nthreads) {
    const int kkh = i >> 6;                      // 0..31 = kk*2 + kh
    const int c   = i & 63;
    const int k0  = (kkh >> 1) * 4 + (kkh & 1) * 2;
    v2f w;
    w.x = W[k0 * 64 + c];
    w.y = W[(k0 + 1) * 64 + c];
    Wl[i] = w;
  }
}

// [16,64] (LDS xs, row stride 64 floats) @ [64,64] (LDS Wl, packed) -> 4 col-tiles.
// A-fragments hoisted to 16 v2f registers, reused across the 4 N-tiles.
__device__ __forceinline__ void gemm_16x64x64(const float* xs, const v2f* Wl,
                                              v8f acc[4], int lane) {
  const int n  = lane & 15;
  const int kh = lane >> 4;
  v2f a[16];
#pragma unroll
  for (int kk = 0; kk < 16; ++kk)
    a[kk] = *(const v2f*)&xs[n * 64 + kk * 4 + 2 * kh];   // ds_load_b64
#pragma unroll
  for (int nt = 0; nt < 4; ++nt) {
    v8f c = {};
#pragma unroll
    for (int kk = 0; kk < 16; ++kk)
      c = wmma_f32(a[kk], Wl[(kk * 2 + kh) * 64 + nt * 16 + n], c);
    acc[nt] = c;
  }
}

// --------------------------------------------------------------------------
// Fused gather -> (x@W + b) via WMMA -> atomic scatter-add into dst segments.
// dst[idxD[t]] += (srcA[idxA[t]] + srcB[idxB[t]]) @ W + b   for t in [0,T)
// One wave handles 16 triangles; 4 waves per block; W cached packed in LDS.
// --------------------------------------------------------------------------
__launch_bounds__(128)
__global__ void k_gather_proj_scatter(const float* __restrict__ srcA,
                                      const float* __restrict__ srcB,
                                      const int* __restrict__ idxA,
                                      const int* __restrict__ idxB,
                                      const int* __restrict__ idxD,
                                      const float* __restrict__ W,
                                      const float* __restrict__ bias,
                                      float* __restrict__ dst, int T) {
  __shared__ v2f   Wl[32 * 64];
  __shared__ float bs[64];
  __shared__ float xs[4][16 * 64];

  const int tid = threadIdx.x;
  pack_W(Wl, W, tid, 128);
  if (tid < 64) bs[tid] = bias[tid];
  __syncthreads();

  const int wave = tid >> 5;
  const int lane = tid & 31;
  const int t0   = (blockIdx.x * 4 + wave) * 16;
  float* x = xs[wave];

  // Gather 16 rows of 64 floats (float2 per lane per row), sum the two sources.
#pragma unroll
  for (int r = 0; r < 16; ++r) {
    const int t = t0 + r;                 // uniform per wave -> EXEC stays full
    v2f v = {};
    if (t < T) {
      const long long ra = (long long)idxA[t] * 64 + lane * 2;
      const long long rb = (long long)idxB[t] * 64 + lane * 2;
      v.x = srcA[ra] + srcB[rb];
      v.y = srcA[ra + 1] + srcB[rb + 1];
    }
    *(v2f*)&x[r * 64 + lane * 2] = v;     // ds_store_b64
  }

  v8f acc[4];
  gemm_16x64x64(x, Wl, acc, lane);

  const int n  = lane & 15;
  const int kh = lane >> 4;
  int di[8];
#pragma unroll
  for (int r = 0; r < 8; ++r) {
    const int t = t0 + r + 8 * kh;        // D row m = r + 8*kh
    di[r] = (t < T) ? idxD[t] : -1;
  }
#pragma unroll
  for (int nt = 0; nt < 4; ++nt) {
    const float bv = bs[nt * 16 + n];
#pragma unroll
    for (int r = 0; r < 8; ++r) {
      if (di[r] >= 0)
        unsafeAtomicAdd(dst + (long long)di[r] * 64 + nt * 16 + n,
                        acc[nt][r] + bv);  // -> global_atomic_add_f32
    }
  }
}

// --------------------------------------------------------------------------
// Fused combine + 2-layer MLP:
// h = base[t] + m2[t] + m2[inv[t]] ;  out[t] = relu(h@W1 + b1)@W2 + b2
// --------------------------------------------------------------------------
__launch_bounds__(128)
__global__ void k_combine_mlp(const float* __restrict__ base,
                              const float* __restrict__ m2,
                              const int* __restrict__ inv,
                              const float* __restrict__ W1,
                              const float* __restrict__ b1,
                              const float* __restrict__ W2,
                              const float* __restrict__ b2,
                              float* __restrict__ out, int N) {
  __shared__ v2f   W1l[32 * 64];
  __shared__ v2f   W2l[32 * 64];
  __shared__ float b1s[64], b2s[64];
  __shared__ float xs[4][16 * 64];

  const int tid = threadIdx.x;
  pack_W(W1l, W1, tid, 128);
  pack_W(W2l, W2, tid, 128);
  if (tid < 64) { b1s[tid] = b1[tid]; b2s[tid] = b2[tid]; }
  __syncthreads();

  const int wave = tid >> 5;
  const int lane = tid & 31;
  const int t0   = (blockIdx.x * 4 + wave) * 16;
  float* x = xs[wave];

#pragma unroll
  for (int r = 0; r < 16; ++r) {
    const int t = t0 + r;
    v2f v = {};
    if (t < N) {
      const long long rp = (long long)t * 64 + lane * 2;
      const long long ri = (long long)inv[t] * 64 + lane * 2;
      v.x = base[rp] + m2[rp] + m2[ri];
      v.y = base[rp + 1] + m2[rp + 1] + m2[ri + 1];
    }
    *(v2f*)&x[r * 64 + lane * 2] = v;
  }

  const int n  = lane & 15;
  const int kh = lane >> 4;

  v8f acc[4];
  gemm_16x64x64(x, W1l, acc, lane);
  // relu(z + b1) back to LDS (each (m,col) written by exactly one lane/reg)
#pragma unroll
  for (int nt = 0; nt < 4; ++nt) {
    const float bv = b1s[nt * 16 + n];
#pragma unroll
    for (int r = 0; r < 8; ++r) {
      float v = acc[nt][r] + bv;
      x[(r + 8 * kh) * 64 + nt * 16 + n] = v > 0.f ? v : 0.f;
    }
  }
  gemm_16x64x64(x, W2l, acc, lane);
#pragma unroll
  for (int nt = 0; nt < 4; ++nt) {
    const float bv = b2s[nt * 16 + n];
#pragma unroll
    for (int r = 0; r < 8; ++r)
      x[(r + 8 * kh) * 64 + nt * 16 + n] = acc[nt][r] + bv;
  }
  // coalesced 8-byte stores
#pragma unroll
  for (int r = 0; r < 16; ++r) {
    const int t = t0 + r;
    if (t < N) {
      const long long rp = (long long)t * 64 + lane * 2;
      *(v2f*)&out[rp] = *(const v2f*)&x[r * 64 + lane * 2];
    }
  }
}

__global__ void k_copy4(float4* __restrict__ dst, const float4* __restrict__ src,
                        long long n4) {
  long long i = (long long)blockIdx.x * blockDim.x + threadIdx.x;
  if (i < n4) dst[i] = src[i];
}

__global__ void k_zero4(float4* __restrict__ dst, long long n4) {
  long long i = (long long)blockIdx.x * blockDim.x + threadIdx.x;
  if (i < n4) dst[i] = make_float4(0.f, 0.f, 0.f, 0.f);
}

extern "C" void kernel_launch(void* const* d_in, const int* in_sizes, int n_in,
                              void* d_out, int out_size, void* d_ws, size_t ws_size,
                              hipStream_t stream) {
  const float* edge_attr  = (const float*)d_in[0];
  const float* edge_attr2 = (const float*)d_in[1];
  const float* Wp  = (const float*)d_in[2];   // [6,64,64]
  const float* bp  = (const float*)d_in[3];   // [6,64]
  const float* W1a = (const float*)d_in[4];
  const float* b1a = (const float*)d_in[5];
  const float* W2a = (const float*)d_in[6];
  const float* b2a = (const float*)d_in[7];
  const float* W1b = (const float*)d_in[8];
  const float* b1b = (const float*)d_in[9];
  const float* W2b = (const float*)d_in[10];
  const float* b2b = (const float*)d_in[11];
  const int* tri111 = (const int*)d_in[12];   // [3,T]: ij, ik, kj
  const int* tri112 = (const int*)d_in[13];
  const int* tri122 = (const int*)d_in[14];
  const int* tri222 = (const int*)d_in[15];
  const int* inv1 = (const int*)d_in[16];
  const int* inv2 = (const int*)d_in[17];

  const int E  = in_sizes[0] / 64;
  const int E2 = in_sizes[1] / 64;
  const int T  = in_sizes[12] / 3;

  // Workspace layout (phase 2 reuses phase 1 region; needs 2*E2*64 floats).
  float* ws      = (float*)d_ws;
  float* accA    = ws;                        // [E,64]  = edge_attr + m111 + m122
  float* accM112 = ws + (long long)E * 64;    // [E,64]  = m112
  float* accB    = ws;                        // [E2,64] = edge_attr2 + m211 + m222
  float* accM212 = ws + (long long)E2 * 64;   // [E2,64] = m212

  float* outA = (float*)d_out;                // new edge_attr  [E,64]
  float* outB = outA + (long long)E * 64;     // new edge_attr2 [E2,64]

  const int THREADS  = 128;                   // 4 waves, 16 rows each
  const int tBlocks  = (T + 63) / 64;
  const int eBlocks  = (E + 63) / 64;
  const int e2Blocks = (E2 + 63) / 64;

  // ---- phase 1: 1-hop edges ----
  const long long nE4 = (long long)E * 16;    // E*64/4 float4s
  k_copy4<<<(int)((nE4 + 255) / 256), 256, 0, stream>>>((float4*)accA, (const float4*)edge_attr, nE4);
  k_zero4<<<(int)((nE4 + 255) / 256), 256, 0, stream>>>((float4*)accM112, nE4);

  // m111: proj0(e1[ik]+e1[kj]) -> ij
  k_gather_proj_scatter<<<tBlocks, THREADS, 0, stream>>>(
      edge_attr, edge_attr, tri111 + T, tri111 + 2 * T, tri111,
      Wp + 0 * 4096, bp + 0 * 64, accA, T);
  // m112: proj1(e1[ik]+e2[kj]) -> ij
  k_gather_proj_scatter<<<tBlocks, THREADS, 0, stream>>>(
      edge_attr, edge_attr2, tri112 + T, tri112 + 2 * T, tri112,
      Wp + 1 * 4096, bp + 1 * 64, accM112, T);
  // m122: proj2(e2[ik]+e2[kj]) -> ij
  k_gather_proj_scatter<<<tBlocks, THREADS, 0, stream>>>(
      edge_attr2, edge_attr2, tri122 + T, tri122 + 2 * T, tri122,
      Wp + 2 * 4096, bp + 2 * 64, accA, T);

  k_combine_mlp<<<eBlocks, THREADS, 0, stream>>>(
      accA, accM112, inv1, W1a, b1a, W2a, b2a, outA, E);

  // ---- phase 2: 2-hop edges (uses updated edge_attr from outA) ----
  const long long nE24 = (long long)E2 * 16;
  k_copy4<<<(int)((nE24 + 255) / 256), 256, 0, stream>>>((float4*)accB, (const float4*)edge_attr2, nE24);
  k_zero4<<<(int)((nE24 + 255) / 256), 256, 0, stream>>>((float4*)accM212, nE24);

  // m211: proj3(e1'[ij112]+e1'[ik112]) -> kj112
  k_gather_proj_scatter<<<tBlocks, THREADS, 0, stream>>>(
      outA, outA, tri112 + 0, tri112 + T, tri112 + 2 * T,
      Wp + 3 * 4096, bp + 3 * 64, accB, T);
  // m212: proj4(e1'[ij122]+e2[kj122]) -> ik122
  k_gather_proj_scatter<<<tBlocks, THREADS, 0, stream>>>(
      outA, edge_attr2, tri122 + 0, tri122 + 2 * T, tri122 + T,
      Wp + 4 * 4096, bp + 4 * 64, accM212, T);
  // m222: proj5(e2[ik]+e2[kj]) -> ij
  k_gather_proj_scatter<<<tBlocks, THREADS, 0, stream>>>(
      edge_attr2, edge_attr2, tri222 + T, tri222 + 2 * T, tri222,
      Wp + 5 * 4096, bp + 5 * 64, accB, T);

  k_combine_mlp<<<e2Blocks, THREADS, 0, stream>>>(
      accB, accM212, inv2, W1b, b1b, W2b, b2b, outB, E2);
}